// CapsuleNet_2173253452541
// MI455X (gfx1250) — compile-verified
//
#include <hip/hip_runtime.h>
#include <math.h>

// CapsuleNet for MI455X (gfx1250, wave32).
// n=20000 nodes, nfeat=500, d=128 (=8 capsules x 16), m=32 neighbors,
// 6 routing iterations x 2 layers, CUT=5, nclass=16.

#define NCAPS  8
#define NHID   16
#define DFEAT  128      // NCAPS*NHID
#define NFEAT  500
#define MNEI   32
#define CUTN   5
#define NCLASS 16
#define ROUTIT 6

typedef __attribute__((ext_vector_type(2))) float v2f;
typedef __attribute__((ext_vector_type(4))) float v4f;
typedef __attribute__((ext_vector_type(8))) float v8f;

// FP32 WMMA: D(16x16) = A(16x4) * B(4x16) + C.
// A layout: lanes 0-15 hold M=lane, K=k0..k0+1 in the 2 VGPRs; lanes 16-31 K=k0+2..k0+3.
// B layout: VGPR0 holds K=k0 (lanes 0-15, N=lane) / K=k0+2 (lanes 16-31); VGPR1 = +1.
// C/D layout: VGPR r holds M=r (lanes 0-15) / M=r+8 (lanes 16-31), N = lane&15.
__device__ __forceinline__ v8f wmma_f32_16x16x4(v2f a, v2f b, v8f c) {
#if defined(__HIP_DEVICE_COMPILE__) && __has_builtin(__builtin_amdgcn_wmma_f32_16x16x4_f32)
  return __builtin_amdgcn_wmma_f32_16x16x4_f32(false, a, false, b, (short)0, c,
                                               false, false);
#else
  // host-pass / missing-builtin placeholder (never the intended device path)
  c[0] += a.x * b.x;
  return c;
#endif
}

// Async 16B global->LDS copy (CDNA5 async path, tracked by ASYNCcnt).
// ldsaddr = low 32 bits of the generic pointer (flat->LDS truncation, ISA 10.2).
__device__ __forceinline__ void async_copy_b128(float* lds_dst,
                                                const float* gsrc) {
#if defined(__HIP_DEVICE_COMPILE__) && defined(__gfx1250__)
  unsigned ldsa = (unsigned)(unsigned long long)lds_dst;
  asm volatile("global_load_async_to_lds_b128 %0, %1, off"
               :
               : "v"(ldsa), "v"(gsrc)
               : "memory");
#else
  *(v4f*)lds_dst = *(const v4f*)gsrc;
#endif
}

__device__ __forceinline__ void async_wait_all() {
#if defined(__HIP_DEVICE_COMPILE__) && defined(__gfx1250__)
  asm volatile("s_wait_asynccnt 0x0" ::: "memory");
#endif
}

// ---------------------------------------------------------------------------
// 1) h = x @ pca_w + pca_b      [n,500] x [500,128] -> [n,128]
//    8 waves/block, each wave owns a 16-row x 128-col tile (8 f32 WMMA accums).
//    pca_w K-tiles staged through LDS in interleaved K-pair layout:
//      Bs[(kr>>1)*256 + col*2 + (kr&1)]
//    so each lane's WMMA B operand (K, K+1 at one column) is a single aligned
//    ds_load_b64 into an even VGPR pair (no v_mov reassembly).
// ---------------------------------------------------------------------------
__global__ __launch_bounds__(256) void pca_wmma_kernel(
    const float* __restrict__ x, const float* __restrict__ w,
    const float* __restrict__ bias, float* __restrict__ h, int n) {
  __shared__ __align__(16) float Bs[16 * DFEAT];  // 8 KB: 16 K-rows of pca_w
  const int tid  = threadIdx.x;
  const int wave = tid >> 5;
  const int lane = tid & 31;
  const int r0   = blockIdx.x * 128 + wave * 16;
  const int hi   = lane >> 4;                    // 0: lanes 0-15, 1: lanes 16-31
  const int col  = lane & 15;
  const int arow = min(r0 + col, n - 1);         // clamped (stores guarded)
  const float* __restrict__ xrow = x + (size_t)arow * NFEAT;

  v8f acc[8];
#pragma unroll
  for (int i = 0; i < 8; ++i) acc[i] = v8f{0, 0, 0, 0, 0, 0, 0, 0};

  for (int kt = 0; kt < NFEAT; kt += 16) {
    __syncthreads();
    for (int i = tid; i < 16 * DFEAT; i += 256) {
      const int kr  = i >> 7;                      // 0..15 within tile
      const int cc  = i & 127;
      const int krg = kt + kr;
      const int krc = min(krg, NFEAT - 1);         // clamped, branchless
      const float val = w[(size_t)krc * DFEAT + cc];
      Bs[(kr >> 1) * 256 + cc * 2 + (kr & 1)] = (krg < NFEAT) ? val : 0.0f;
    }
    __syncthreads();
    if (kt + 16 < NFEAT)  // prefetch next x K-tile (global_prefetch_b8)
      __builtin_prefetch(xrow + kt + 16, 0, 3);
#pragma unroll
    for (int k4 = 0; k4 < 16; k4 += 4) {
      const int kk  = kt + k4 + hi * 2;
      const int kkc = min(kk, NFEAT - 2);          // even, 8B-aligned
      v2f a = *(const v2f*)(xrow + kkc);
      const bool ok = kk < NFEAT;
      a.x = ok ? a.x : 0.0f;
      a.y = ok ? a.y : 0.0f;
      const int pr = (k4 >> 1) + hi;               // K-pair row (krow>>1)
#pragma unroll
      for (int nb8 = 0; nb8 < 8; ++nb8) {
        const v2f b = *(const v2f*)&Bs[pr * 256 + (nb8 * 16 + col) * 2];
        acc[nb8] = wmma_f32_16x16x4(a, b, acc[nb8]);
      }
    }
  }

  if (r0 < n) {  // n % 16 == 0 -> full-tile-or-nothing
    const int rbase = r0 + hi * 8;
#pragma unroll
    for (int nb8 = 0; nb8 < 8; ++nb8) {
      const float bv = bias[nb8 * 16 + col];
#pragma unroll
      for (int r = 0; r < 8; ++r)
        h[(size_t)(rbase + r) * DFEAT + nb8 * 16 + col] = acc[nb8][r] + bv;
    }
  }
}

// ---------------------------------------------------------------------------
// 2) attention loss: per-node LN(q) @ w_qs, h3 @ w_ks, softmax(q.k/4),
//    loss += 1 - p[c1][c1]. Block partials -> ws, reduced by attn_reduce.
// ---------------------------------------------------------------------------
__global__ __launch_bounds__(256) void attn_kernel(
    const float* __restrict__ h, const float* __restrict__ ln_g,
    const float* __restrict__ ln_b, const float* __restrict__ wqs,
    const float* __restrict__ wks, float* __restrict__ partial, int n) {
  __shared__ float sWq[NHID * NHID], sWk[NHID * NHID], sG[NHID], sB[NHID];
  __shared__ float red[256];
  const int tid = threadIdx.x;
  if (tid < NHID * NHID) { sWq[tid] = wqs[tid]; sWk[tid] = wks[tid]; }
  if (tid < NHID) { sG[tid] = ln_g[tid]; sB[tid] = ln_b[tid]; }
  __syncthreads();

  const int v = blockIdx.x * 256 + tid;
  float loss = 0.0f;
  if (v < n) {
    const float* hv = h + (size_t)v * DFEAT;
    float kk[NCAPS][NHID];
    for (int c2 = 0; c2 < NCAPS; ++c2) {
      float hr[NHID];
#pragma unroll
      for (int d = 0; d < NHID; ++d) hr[d] = hv[c2 * NHID + d];
#pragma unroll
      for (int j = 0; j < NHID; ++j) {
        float s = 0.0f;
#pragma unroll
        for (int d = 0; d < NHID; ++d) s += hr[d] * sWk[d * NHID + j];
        kk[c2][j] = s;
      }
    }
    for (int c1 = 0; c1 < NCAPS; ++c1) {
      float hr[NHID], qn[NHID], q[NHID];
      float mu = 0.0f;
#pragma unroll
      for (int d = 0; d < NHID; ++d) { hr[d] = hv[c1 * NHID + d]; mu += hr[d]; }
      mu *= (1.0f / NHID);
      float var = 0.0f;
#pragma unroll
      for (int d = 0; d < NHID; ++d) { float t = hr[d] - mu; var += t * t; }
      var *= (1.0f / NHID);
      const float inv = 1.0f / sqrtf(var + 1e-6f);
#pragma unroll
      for (int d = 0; d < NHID; ++d) qn[d] = (hr[d] - mu) * inv * sG[d] + sB[d];
#pragma unroll
      for (int j = 0; j < NHID; ++j) {
        float s = 0.0f;
#pragma unroll
        for (int d = 0; d < NHID; ++d) s += qn[d] * sWq[d * NHID + j];
        q[j] = s;
      }
      float logit[NCAPS];
      float mx = -3.0e38f;
      for (int c2 = 0; c2 < NCAPS; ++c2) {
        float s = 0.0f;
#pragma unroll
        for (int j = 0; j < NHID; ++j) s += q[j] * kk[c2][j];
        logit[c2] = 0.25f * s;  // / sqrt(16)
        mx = fmaxf(mx, logit[c2]);
      }
      float se = 0.0f;
      for (int c2 = 0; c2 < NCAPS; ++c2) se += expf(logit[c2] - mx);
      loss += 1.0f - expf(logit[c1] - mx) / se;  // sum of off-diagonal softmax
    }
  }
  red[tid] = loss;
  __syncthreads();
  for (int s = 128; s > 0; s >>= 1) {
    if (tid < s) red[tid] += red[tid + s];
    __syncthreads();
  }
  if (tid == 0) partial[blockIdx.x] = red[0];
}

__global__ void attn_reduce_kernel(const float* __restrict__ partial, int nblocks,
                                   float* __restrict__ out_loss, int n) {
  __shared__ float red[256];
  float s = 0.0f;
  for (int i = threadIdx.x; i < nblocks; i += 256) s += partial[i];
  red[threadIdx.x] = s;
  __syncthreads();
  for (int st = 128; st > 0; st >>= 1) {
    if (threadIdx.x < st) red[threadIdx.x] += red[threadIdx.x + st];
    __syncthreads();
  }
  if (threadIdx.x == 0) out_loss[0] = red[0] / (56.0f * (float)n);
}

// ---------------------------------------------------------------------------
// 3) xn = normalize_per_capsule(relu(in)); row n is the zero "dummy" row.
//    One thread per (node, capsule): 64B contiguous read/write.
// ---------------------------------------------------------------------------
__global__ void norm_relu_kernel(const float* __restrict__ in,
                                 float* __restrict__ xn, int n) {
  const int idx = blockIdx.x * blockDim.x + threadIdx.x;
  if (idx >= (n + 1) * NCAPS) return;
  const int v = idx >> 3, kc = idx & 7;
  float* o = xn + (size_t)v * DFEAT + kc * NHID;
  if (v == n) {
#pragma unroll
    for (int c = 0; c < NHID; ++c) o[c] = 0.0f;
    return;
  }
  const float* pi = in + (size_t)v * DFEAT + kc * NHID;
  float val[NHID];
  float ss = 0.0f;
#pragma unroll
  for (int c = 0; c < NHID; ++c) {
    val[c] = fmaxf(pi[c], 0.0f);
    ss += val[c] * val[c];
  }
  const float inv = 1.0f / fmaxf(sqrtf(ss), 1e-12f);
#pragma unroll
  for (int c = 0; c < NHID; ++c) o[c] = val[c] * inv;
}

// ---------------------------------------------------------------------------
// 4) Dynamic routing: one 256-thread workgroup per node. Neighbor tile
//    (32 x 128 f32 = 16 KB) staged into LDS with global_load_async_to_lds_b128
//    (ASYNCcnt-tracked, no VGPR round trip); 6 iterations in LDS.
//    k=8 softmax via shfl_xor {1,2,4}; dd=16 L2-norm via shfl_xor {1,2,4,8}.
//    If last: store argmax-capsule of final p for the first CUT neighbor slots.
// ---------------------------------------------------------------------------
__global__ __launch_bounds__(256) void routing_kernel(
    const float* __restrict__ xn, const long long* __restrict__ nb,
    float* __restrict__ uout, int* __restrict__ argcaps, int n, int last) {
  __shared__ __align__(16) float z[MNEI * DFEAT];  // 16 KB neighbor features
  __shared__ float x3[DFEAT];
  __shared__ float u[DFEAT];
  __shared__ float p[MNEI * NCAPS];                // 256 floats
  __shared__ int   nbm[MNEI];
  const int tid = threadIdx.x;
  const int v   = blockIdx.x;

  if (tid < DFEAT) x3[tid] = xn[(size_t)v * DFEAT + tid];
  if (tid < MNEI) {
    const long long t = nb[(size_t)v * MNEI + tid];
    nbm[tid] = (t < 0) ? n : (int)t;
  }
  __syncthreads();
  // Async gather: 1024 x 16B; each thread issues 4 async-to-LDS loads.
  for (int i4 = tid; i4 < MNEI * (DFEAT / 4); i4 += 256) {
    const int mm = i4 >> 5;   // neighbor slot (32 float4 per 128-float row)
    const int c4 = i4 & 31;
    async_copy_b128(&z[i4 * 4], xn + (size_t)nbm[mm] * DFEAT + c4 * 4);
  }
  async_wait_all();   // this wave's ASYNCcnt == 0
  __syncthreads();    // all waves' LDS writes visible

  const int m   = tid >> 3;   // neighbor slot for p
  const int kc8 = tid & 7;    // capsule for p
  const int kc  = tid >> 4;   // capsule for u (tid < 128)

  for (int it = 0; it < ROUTIT; ++it) {
    if (it == 0) {
      p[tid] = 0.125f;        // softmax of zeros over k=8
    } else {
      float s = 0.0f;
#pragma unroll
      for (int c = 0; c < NHID; ++c)
        s += z[m * DFEAT + kc8 * NHID + c] * u[kc8 * NHID + c];
      float mx = s;
#pragma unroll
      for (int off = 1; off < 8; off <<= 1) mx = fmaxf(mx, __shfl_xor(mx, off));
      const float e = expf(s - mx);
      float se = e;
#pragma unroll
      for (int off = 1; off < 8; off <<= 1) se += __shfl_xor(se, off);
      p[tid] = e / se;
    }
    __syncthreads();
    if (tid < DFEAT) {
      float acc = x3[tid];
#pragma unroll
      for (int mm = 0; mm < MNEI; ++mm)
        acc += z[mm * DFEAT + tid] * p[mm * NCAPS + kc];
      if (it < ROUTIT - 1) {
        float ss = acc * acc;
#pragma unroll
        for (int off = 1; off < 16; off <<= 1) ss += __shfl_xor(ss, off);
        u[tid] = acc / fmaxf(sqrtf(ss), 1e-12f);
      } else {
        u[tid] = acc;  // final u left unnormalized
      }
    }
    __syncthreads();
  }

  if (tid < DFEAT) uout[(size_t)v * DFEAT + tid] = u[tid];
  if (last && tid < CUTN) {
    float best = p[tid * NCAPS];
    int bi = 0;
#pragma unroll
    for (int kci = 1; kci < NCAPS; ++kci) {
      const float pv = p[tid * NCAPS + kci];
      if (pv > best) { best = pv; bi = kci; }  // strict > -> first max (argmax)
    }
    argcaps[(size_t)v * CUTN + tid] = bi;
  }
}

// ---------------------------------------------------------------------------
// 5) meta: the reference two-hop sum over j collapses (exactly one non-dummy
//    capsule j = argcaps[w1][b] per second hop), so:
//    meta[v,i,c] = u[v,i,c] + 1/25 * sum_{a<5,b<5, argcaps[v][a]==i}
//                  xn[nb[w1][b]][argcaps[w1][b]][c],  w1 = nb[v][a].
//    One thread per (node, c).
// ---------------------------------------------------------------------------
__global__ void meta_kernel(const float* __restrict__ u,
                            const float* __restrict__ xn,
                            const long long* __restrict__ nb,
                            const int* __restrict__ argcaps,
                            float* __restrict__ meta, int n) {
  const int idx = blockIdx.x * blockDim.x + threadIdx.x;
  if (idx >= n * NHID) return;
  const int v = idx >> 4, c = idx & 15;
  float acc[NCAPS];
#pragma unroll
  for (int i = 0; i < NCAPS; ++i) acc[i] = 0.0f;
  for (int a = 0; a < CUTN; ++a) {
    const long long w1l = nb[(size_t)v * MNEI + a];
    if (w1l < 0) continue;  // masked neighbor -> dummy first hop -> zeros
    const int w1 = (int)w1l;
    const int i  = argcaps[(size_t)v * CUTN + a];
    float s = 0.0f;
    for (int b = 0; b < CUTN; ++b) {
      const long long w2l = nb[(size_t)w1 * MNEI + b];
      const int j  = argcaps[(size_t)w1 * CUTN + b];
      const int w2 = (w2l < 0) ? n : (int)w2l;  // row n is all-zero
      s += xn[(size_t)w2 * DFEAT + j * NHID + c];
    }
    acc[i] += s;
  }
  const float inv25 = 1.0f / (float)(CUTN * CUTN);
#pragma unroll
  for (int i = 0; i < NCAPS; ++i)
    meta[(size_t)v * DFEAT + i * NHID + c] =
        u[(size_t)v * DFEAT + i * NHID + c] + acc[i] * inv25;
}

// ---------------------------------------------------------------------------
// 6) out = relu(meta) @ mlp_w + mlp_b, then row log-softmax (across the 16
//    lanes of each wave32 half via shfl_xor). WMMA f32, 16 rows per wave.
//    mlp_w staged in the same interleaved K-pair layout as the PCA kernel.
// ---------------------------------------------------------------------------
__global__ __launch_bounds__(256) void mlp_wmma_kernel(
    const float* __restrict__ meta, const float* __restrict__ w,
    const float* __restrict__ bias, float* __restrict__ out_ls,
    float* __restrict__ out_meta, int n) {
  __shared__ __align__(16) float Bs[DFEAT * NCLASS];  // 8 KB: full mlp_w
  const int tid = threadIdx.x;
  for (int i = tid; i < DFEAT * NCLASS; i += 256) {
    const int kr = i >> 4, cc = i & 15;
    Bs[(kr >> 1) * 32 + cc * 2 + (kr & 1)] = w[i];
  }
  __syncthreads();

  const int wave = tid >> 5, lane = tid & 31;
  const int r0   = blockIdx.x * 128 + wave * 16;
  const int hi   = lane >> 4;
  const int col  = lane & 15;
  const int arow = min(r0 + col, n - 1);

  v8f acc = v8f{0, 0, 0, 0, 0, 0, 0, 0};
#pragma unroll
  for (int k4 = 0; k4 < DFEAT; k4 += 4) {
    const int kk = k4 + hi * 2;
    v2f a;
    a.x = fmaxf(meta[(size_t)arow * DFEAT + kk], 0.0f);
    a.y = fmaxf(meta[(size_t)arow * DFEAT + kk + 1], 0.0f);
    const v2f b = *(const v2f*)&Bs[((k4 >> 1) + hi) * 32 + col * 2];
    acc = wmma_f32_16x16x4(a, b, acc);
  }

  if (r0 < n) {
    const float bv = bias[col];
#pragma unroll
    for (int r = 0; r < 8; ++r) {
      const int row = r0 + hi * 8 + r;
      const float val = acc[r] + bv;
      float mx = val;
#pragma unroll
      for (int off = 1; off < 16; off <<= 1) mx = fmaxf(mx, __shfl_xor(mx, off));
      const float e = expf(val - mx);
      float se = e;
#pragma unroll
      for (int off = 1; off < 16; off <<= 1) se += __shfl_xor(se, off);
      out_ls[(size_t)row * NCLASS + col]   = val - (mx + logf(se));
      out_meta[(size_t)row * NCLASS + col] = val;
    }
  }
}

// ---------------------------------------------------------------------------
// Host launcher. Inputs: x, nb(int64), pca_w, pca_b, ln_g, ln_b, w_qs, w_ks,
// mlp_w, mlp_b. Output: [log_softmax n*16][attn_loss 1][meta n*16].
// ---------------------------------------------------------------------------
extern "C" void kernel_launch(void* const* d_in, const int* in_sizes, int n_in,
                              void* d_out, int out_size, void* d_ws, size_t ws_size,
                              hipStream_t stream) {
  const float*     x     = (const float*)d_in[0];
  const long long* nb    = (const long long*)d_in[1];
  const float*     pca_w = (const float*)d_in[2];
  const float*     pca_b = (const float*)d_in[3];
  const float*     ln_g  = (const float*)d_in[4];
  const float*     ln_b  = (const float*)d_in[5];
  const float*     w_qs  = (const float*)d_in[6];
  const float*     w_ks  = (const float*)d_in[7];
  const float*     mlp_w = (const float*)d_in[8];
  const float*     mlp_b = (const float*)d_in[9];
  (void)n_in; (void)out_size; (void)ws_size;

  const int n = in_sizes[1] / MNEI;  // 20000

  float* out_ls   = (float*)d_out;
  float* out_loss = out_ls + (size_t)n * NCLASS;
  float* out_meta = out_loss + 1;

  // workspace carve-out (256B aligned): hbuf | xnbuf(+dummy row) | metabuf |
  // argcaps | partials    (~31 MB total)
  char* wsb = (char*)d_ws;
  size_t off = 0;
  auto carve = [&](size_t bytes) {
    void* p = wsb + off;
    off = (off + bytes + 255) & ~(size_t)255;
    return p;
  };
  float* hbuf    = (float*)carve((size_t)n * DFEAT * sizeof(float));
  float* xnbuf   = (float*)carve((size_t)(n + 1) * DFEAT * sizeof(float));
  float* metabuf = (float*)carve((size_t)n * DFEAT * sizeof(float));
  int*   argcaps = (int*)carve((size_t)n * CUTN * sizeof(int));
  float* partial = (float*)carve(1024 * sizeof(float));

  const int gridGemm = (n + 127) / 128;
  pca_wmma_kernel<<<gridGemm, 256, 0, stream>>>(x, pca_w, pca_b, hbuf, n);

  const int gridAttn = (n + 255) / 256;
  attn_kernel<<<gridAttn, 256, 0, stream>>>(hbuf, ln_g, ln_b, w_qs, w_ks,
                                            partial, n);
  attn_reduce_kernel<<<1, 256, 0, stream>>>(partial, gridAttn, out_loss, n);

  const int gridNorm = ((n + 1) * NCAPS + 255) / 256;
  // layer 0: xn = norm(relu(h)); u0 -> hbuf (raw; relu happens in next norm)
  norm_relu_kernel<<<gridNorm, 256, 0, stream>>>(hbuf, xnbuf, n);
  routing_kernel<<<n, 256, 0, stream>>>(xnbuf, nb, hbuf, argcaps, n, 0);
  // layer 1 (last): xn = norm(relu(u0)); u1 -> hbuf, argcaps from final p
  norm_relu_kernel<<<gridNorm, 256, 0, stream>>>(hbuf, xnbuf, n);
  routing_kernel<<<n, 256, 0, stream>>>(xnbuf, nb, hbuf, argcaps, n, 1);

  const int gridMeta = (n * NHID + 255) / 256;
  meta_kernel<<<gridMeta, 256, 0, stream>>>(hbuf, xnbuf, nb, argcaps, metabuf, n);

  mlp_wmma_kernel<<<gridGemm, 256, 0, stream>>>(metabuf, mlp_w, mlp_b, out_ls,
                                                out_meta, n);
}